// app_tem_42700564857124
// MI455X (gfx1250) — compile-verified
//
#include <hip/hip_runtime.h>

typedef __attribute__((ext_vector_type(2))) float v2f;
typedef __attribute__((ext_vector_type(4))) float v4f;
typedef __attribute__((ext_vector_type(8))) float v8f;

#define TT      16      // frames per batch
#define DD      2048    // feature dim
#define TKEEP   15
#define ROWPAD  2052    // 2048 + 4 floats: kills LDS bank conflicts for WMMA-layout reads
#define NWAVES  8
#define BLOCK   256
#define EPS_F   1e-8f

// Direct HBM -> LDS async copy (gfx1250 GLOBAL_LOAD_ASYNC_TO_LDS_B128, GVS mode).
// lds_off: wave-relative LDS byte offset (low 32 bits of generic LDS pointer)
// gbyte  : per-lane 32-bit byte offset added to the 64-bit SGPR base
__device__ __forceinline__ void async_copy_b128(unsigned lds_off,
                                                unsigned gbyte,
                                                const float* __restrict__ base) {
  asm volatile("global_load_async_to_lds_b128 %0, %1, %2 th:TH_LOAD_NT"
               :: "v"(lds_off), "v"(gbyte), "s"(base)
               : "memory");
}

__device__ __forceinline__ void wait_async_zero() {
  asm volatile("s_wait_asynccnt 0x0" ::: "memory");
}

__global__ __launch_bounds__(BLOCK) void app_tem_42700564857124_kernel(
    const float* __restrict__ in, float* __restrict__ out) {
  __shared__ float xs[TT * ROWPAD];          // staged batch tile (row-padded)
  __shared__ float gpart[NWAVES * 256];      // per-wave partial Gram matrices
  __shared__ float gram[256];                // reduced 16x16 Gram
  __shared__ float inv_norm[TT];
  __shared__ float safe_n[TT];
  __shared__ float score[TT];
  __shared__ int   remove_id;

  const int tid = threadIdx.x;
  const int b   = blockIdx.x;
  const float* __restrict__ src = in + (size_t)b * (TT * DD);

  // ---- Stage X[16][2048] into LDS via async DMA (no VGPR round-trip) ----
  // 8192 x 128-bit transfers; 32 wave-wide async instructions per wave (< ASYNCcnt max 63).
  for (int i = tid; i < (TT * DD / 4); i += BLOCK) {
    const int row = i >> 9;              // 512 x 128-bit per row
    const int c4  = i & 511;
    const unsigned lds_off = (unsigned)(uintptr_t)&xs[row * ROWPAD + c4 * 4];
    async_copy_b128(lds_off, (unsigned)i * 16u, src);
  }
  wait_async_zero();                     // this wave's transfers landed in LDS
  __syncthreads();                       // make all waves' tiles visible

  // ---- Gram = X * X^T via V_WMMA_F32_16X16X4_F32 (A and B share registers) ----
  const int lane  = tid & 31;
  const int wave  = tid >> 5;
  const int m     = lane & 15;           // A row / B col for this lane
  const int khalf = lane >> 4;           // 0: K={0,1}, 1: K={2,3} within a step
  const int kbase0 = wave * (DD / NWAVES);

  v8f acc = {};
  #pragma unroll 4
  for (int k = 0; k < DD / NWAVES; k += 4) {
    const int col = kbase0 + k + 2 * khalf;
    v2f a = *(const v2f*)&xs[m * ROWPAD + col];   // ds_load_b64, conflict-free
    acc = __builtin_amdgcn_wmma_f32_16x16x4_f32(
        /*neg_a=*/false, a, /*neg_b=*/false, a,
        /*c_mod=*/(short)0, acc, /*reuse_a=*/false, /*reuse_b=*/false);
  }

  // Spill partial Gram: C/D layout -> element (row = v + 8*khalf, col = m)
  #pragma unroll
  for (int v = 0; v < 8; ++v) {
    const int row = v + 8 * khalf;
    gpart[wave * 256 + row * 16 + m] = acc[v];
  }
  __syncthreads();

  // ---- Deterministic cross-wave reduction (fixed summation order) ----
  {
    float s = 0.f;
    #pragma unroll
    for (int w = 0; w < NWAVES; ++w) s += gpart[w * 256 + tid];
    gram[tid] = s;
  }
  __syncthreads();

  // ---- Norms (diag), cosine row-sums, first-index argmin ----
  if (tid < TT) {
    const float n = sqrtf(gram[tid * 16 + tid]);
    inv_norm[tid] = 1.0f / n;              // reference's final normalize: no eps clamp
    safe_n[tid]   = fmaxf(n, EPS_F);       // eps-clamped norm for cosine sim
  }
  __syncthreads();
  if (tid < TT) {
    float si = 0.f;
    #pragma unroll
    for (int j = 0; j < TT; ++j) si += gram[tid * 16 + j] / safe_n[j];
    score[tid] = si / safe_n[tid];
  }
  __syncthreads();
  if (tid == 0) {
    int best = 0;
    float bv = score[0];
    #pragma unroll
    for (int i = 1; i < TT; ++i) {
      if (score[i] < bv) { bv = score[i]; best = i; }   // strict <: first min wins
    }
    remove_id = best;
  }
  __syncthreads();

  // ---- Stream out 15 kept rows, scaled by 1/norm, non-temporal stores ----
  const int rid = remove_id;
  float* __restrict__ dst = out + (size_t)b * (TKEEP * DD);
  for (int i = tid; i < (TKEEP * DD / 4); i += BLOCK) {
    const int r    = i >> 9;               // 512 x 128-bit per output row
    const int c4   = i & 511;
    const int keep = r + (r >= rid ? 1 : 0);
    v4f v = *(const v4f*)&xs[keep * ROWPAD + c4 * 4];
    v *= inv_norm[keep];
    __builtin_nontemporal_store(v, (v4f*)(dst + (size_t)i * 4));
  }
}

extern "C" void kernel_launch(void* const* d_in, const int* in_sizes, int n_in,
                              void* d_out, int out_size, void* d_ws, size_t ws_size,
                              hipStream_t stream) {
  const float* in  = (const float*)d_in[0];
  float*       out = (float*)d_out;
  const int B = in_sizes[0] / (TT * DD);   // 4096
  app_tem_42700564857124_kernel<<<dim3(B), dim3(BLOCK), 0, stream>>>(in, out);
}